// PurityBCE_34565896798884
// MI455X (gfx1250) — compile-verified
//
#include <hip/hip_runtime.h>

// Problem constants (from reference: [64, 512, 512] float32 images)
#define N_IMG 64
#define H_IMG 512
#define W_IMG 512
// Tile: 256 wide x 16 tall outputs per 256-thread block
#define TW 256
#define TH 16
#define LROWS (TH + 2)        // 18 input rows (with y halo)
#define LSTRIDE 264           // floats per LDS row; bulk region starts at lx=4 -> 16B aligned

// ---- CDNA5 async global->LDS copies (ASYNCcnt path) -------------------------
__device__ __forceinline__ void async_load_f4(unsigned lds_byte_addr, const float* gp) {
    // LDS dest address in VGPR (byte offset in wave's LDS), 64-bit global address in VGPR pair
    asm volatile("global_load_async_to_lds_b128 %0, %1, off"
                 :: "v"(lds_byte_addr), "v"(gp) : "memory");
}
__device__ __forceinline__ void async_load_f1(unsigned lds_byte_addr, const float* gp) {
    asm volatile("global_load_async_to_lds_b32 %0, %1, off"
                 :: "v"(lds_byte_addr), "v"(gp) : "memory");
}
__device__ __forceinline__ void wait_async0() {
    asm volatile("s_wait_asynccnt 0" ::: "memory");
}

__global__ __launch_bounds__(256)
void purity_bce_main(const float* __restrict__ outI,
                     const float* __restrict__ segI,
                     float* __restrict__ partial) {
    __shared__ __attribute__((aligned(16))) float sO[LROWS * LSTRIDE];
    __shared__ __attribute__((aligned(16))) float sS[LROWS * LSTRIDE];
    __shared__ float red[256];

    const int tid = threadIdx.x;
    const int x0 = blockIdx.x * TW;
    const int y0 = blockIdx.y * TH;
    const int n  = blockIdx.z;
    const size_t img_base = (size_t)n * H_IMG * W_IMG;

    // 1) Zero both LDS tiles: provides the zero padding for all OOB halo entries.
    float4 z4 = make_float4(0.f, 0.f, 0.f, 0.f);
    for (int i = tid; i < (LROWS * LSTRIDE) / 4; i += 256) {
        ((float4*)sO)[i] = z4;
        ((float4*)sS)[i] = z4;
    }
    __syncthreads();  // zeros committed before async engine writes LDS

    // Low 32 bits of a generic pointer to __shared__ == LDS byte offset (aperture in high bits)
    const unsigned baseO = (unsigned)(size_t)(void*)sO;
    const unsigned baseS = (unsigned)(size_t)(void*)sS;

    // 2) Bulk async load: LROWS rows x 64 float4 (covers global x0..x0+255 at lx=4..259)
    for (int i = tid; i < LROWS * 64; i += 256) {
        const int ly = i >> 6;
        const int k  = i & 63;
        const int gy = y0 - 1 + ly;
        if (gy >= 0 && gy < H_IMG) {
            const size_t gofs  = img_base + (size_t)gy * W_IMG + x0 + 4 * k;
            const unsigned lofs = (unsigned)((ly * LSTRIDE + 4 + 4 * k) * sizeof(float));
            async_load_f4(baseO + lofs, outI + gofs);
            async_load_f4(baseS + lofs, segI + gofs);
        }
    }
    // 3) x-halo columns: lx=3 <- gx=x0-1, lx=TW+4 <- gx=x0+TW (b32, EXEC-masked at borders)
    for (int i = tid; i < LROWS * 2; i += 256) {
        const int ly   = i >> 1;
        const int side = i & 1;
        const int gy = y0 - 1 + ly;
        const int gx = side ? (x0 + TW) : (x0 - 1);
        if (gy >= 0 && gy < H_IMG && gx >= 0 && gx < W_IMG) {
            const size_t gofs  = img_base + (size_t)gy * W_IMG + gx;
            const unsigned lofs = (unsigned)((ly * LSTRIDE + (side ? (TW + 4) : 3)) * sizeof(float));
            async_load_f1(baseO + lofs, outI + gofs);
            async_load_f1(baseS + lofs, segI + gofs);
        }
    }
    wait_async0();      // each wave drains its own ASYNCcnt
    __syncthreads();    // tiles visible to all waves

    // 4) Separable stencil out of LDS: thread owns column lx, rolls 3 horizontal sums.
    const int lx = tid + 4;
    float acc = 0.f;

    float hOm, hOc, hOp, cOc, cOp;
    float hSm, hSc, hSp, cSc, cSp;
    {
        const float* r0 = sO + 0 * LSTRIDE + lx;
        hOm = r0[-1] + r0[0] + r0[1];
        const float* q0 = sS + 0 * LSTRIDE + lx;
        hSm = q0[-1] + q0[0] + q0[1];
        const float* r1 = sO + 1 * LSTRIDE + lx;
        cOc = r1[0]; hOc = r1[-1] + cOc + r1[1];
        const float* q1 = sS + 1 * LSTRIDE + lx;
        cSc = q1[0]; hSc = q1[-1] + cSc + q1[1];
    }
    #pragma unroll 4
    for (int r = 0; r < TH; ++r) {
        const float* r2 = sO + (r + 2) * LSTRIDE + lx;
        cOp = r2[0]; hOp = r2[-1] + cOp + r2[1];
        const float* q2 = sS + (r + 2) * LSTRIDE + lx;
        cSp = q2[0]; hSp = q2[-1] + cSp + q2[1];

        const float o1 = hOm + hOc + hOp - cOc;   // 8-neighbor sum of out
        const float s1 = hSm + hSc + hSp - cSc;   // 8-neighbor sum of seg

        const float po = fmaf(o1,  0.125f, 1e-5f);          // o1/8 + eps
        const float qo = fmaf(o1, -0.125f, 1.0f + 1e-5f);   // (8-o1)/8 + eps
        const float ps = s1 * 0.125f;
        acc -= ps * __logf(po) + (1.0f - ps) * __logf(qo);

        hOm = hOc; hOc = hOp; cOc = cOp;
        hSm = hSc; hSc = hSp; cSc = cSp;
    }

    // 5) Deterministic in-block tree reduction
    red[tid] = acc;
    __syncthreads();
    for (int s = 128; s > 0; s >>= 1) {
        if (tid < s) red[tid] += red[tid + s];
        __syncthreads();
    }
    if (tid == 0) {
        const int bid = blockIdx.x + gridDim.x * (blockIdx.y + gridDim.y * blockIdx.z);
        partial[bid] = red[0];
    }
}

__global__ __launch_bounds__(256)
void purity_bce_reduce(const float* __restrict__ partial, int nblocks,
                       float* __restrict__ out) {
    __shared__ float red[256];
    const int tid = threadIdx.x;
    float a = 0.f;
    for (int i = tid; i < nblocks; i += 256) a += partial[i];  // fixed order per thread
    red[tid] = a;
    __syncthreads();
    for (int s = 128; s > 0; s >>= 1) {
        if (tid < s) red[tid] += red[tid + s];
        __syncthreads();
    }
    if (tid == 0)
        out[0] = red[0] * (1.0f / ((float)N_IMG * (float)H_IMG * (float)W_IMG));
}

extern "C" void kernel_launch(void* const* d_in, const int* in_sizes, int n_in,
                              void* d_out, int out_size, void* d_ws, size_t ws_size,
                              hipStream_t stream) {
    const float* outI = (const float*)d_in[0];   // out_image   [64,512,512] f32
    const float* segI = (const float*)d_in[1];   // segment_image
    float* partial = (float*)d_ws;               // 4096 floats of scratch
    float* out = (float*)d_out;                  // scalar f32

    dim3 grid(W_IMG / TW, H_IMG / TH, N_IMG);    // (2, 32, 64) = 4096 blocks
    purity_bce_main<<<grid, 256, 0, stream>>>(outI, segI, partial);
    const int nblocks = grid.x * grid.y * grid.z;
    purity_bce_reduce<<<1, 256, 0, stream>>>(partial, nblocks, out);
}